// SVDDecomposeTransMatrix_13864154432389
// MI455X (gfx1250) — compile-verified
//
#include <hip/hip_runtime.h>

typedef __attribute__((ext_vector_type(2))) float v2f;
typedef __attribute__((ext_vector_type(8))) float v8f;

// ---------------------------------------------------------------------------
// Setup: cayley(U), cayley(V), M = Cu @ diag @ Cv, store fragment-packed
// ---------------------------------------------------------------------------
__device__ void cayley64(const float* __restrict__ U, float* aug /*64x132*/,
                         float* Q /*64x64*/, int tid) {
  const int S = 132;
  for (int idx = tid; idx < 4096; idx += 256) {
    int i = idx >> 6, j = idx & 63;
    float a = 0.f;
    if (i > j)      a =  U[i * 64 + j];
    else if (i < j) a = -U[j * 64 + i];
    float h = 0.5f * a;
    float d = (i == j) ? 1.f : 0.f;
    aug[i * S + j]      = d - h;   // I - A/2
    aug[i * S + 64 + j] = d + h;   // I + A/2
  }
  __syncthreads();
  __shared__ float fvec[64];
  __shared__ float pivinv;
  for (int k = 0; k < 64; ++k) {
    if (tid == 0) pivinv = 1.0f / aug[k * S + k];
    __syncthreads();
    if (tid < 128) aug[k * S + tid] *= pivinv;
    __syncthreads();
    if (tid < 64) fvec[tid] = aug[tid * S + k];
    __syncthreads();
    for (int idx = tid; idx < 8192; idx += 256) {
      int r = idx >> 7, j = idx & 127;
      if (r != k) aug[r * S + j] -= fvec[r] * aug[k * S + j];
    }
    __syncthreads();
  }
  for (int idx = tid; idx < 4096; idx += 256) {
    int i = idx >> 6, j = idx & 63;
    Q[idx] = aug[i * S + 64 + j];
  }
  __syncthreads();
}

__global__ __launch_bounds__(256) void setup_kernel(
    const float* __restrict__ uL, const float* __restrict__ vL,
    const float* __restrict__ dL, const float* __restrict__ uR,
    const float* __restrict__ vR, const float* __restrict__ dR,
    float* __restrict__ wsA, float* __restrict__ wsB) {
  extern __shared__ float sm[];
  float* aug = sm;            // 64*132 = 8448 floats
  float* Cu  = sm + 8448;     // 4096
  float* Cv  = sm + 12544;    // 4096
  int tid = threadIdx.x;
  bool left = (blockIdx.x == 0);
  const float* U = left ? uL : uR;
  const float* V = left ? vL : vR;
  const float* D = left ? dL : dR;

  cayley64(U, aug, Cu, tid);
  cayley64(V, aug, Cv, tid);

  __shared__ float dvec[64];
  if (tid < 64) dvec[tid] = D[tid];
  __syncthreads();

  float* M = aug;  // reuse: M[i][j] = sum_k Cu[i][k] * d[k] * Cv[k][j]
  for (int idx = tid; idx < 4096; idx += 256) {
    int i = idx >> 6, j = idx & 63;
    float s = 0.f;
    for (int k = 0; k < 64; ++k) s += Cu[i * 64 + k] * dvec[k] * Cv[k * 64 + j];
    M[idx] = s;
  }
  __syncthreads();

  if (left) {
    // Pack MLT = M^T as WMMA A-fragments: frag(mi,k), lane-major, v in {0,1}
    for (int e = tid; e < 4096; e += 256) {
      int frag = e >> 6, mi = frag >> 4, k = frag & 15;
      int le = e & 63, lane = le >> 1, v = le & 1;
      int r, c;
      if (lane < 16) { r = mi * 16 + lane;      c = k * 4 + v;     }
      else           { r = mi * 16 + lane - 16; c = k * 4 + 2 + v; }
      wsA[e] = M[c * 64 + r];   // MLT[r][c] = M[c][r]
    }
  } else {
    // Pack MR = M as WMMA B-fragments: frag(k,nj)
    for (int e = tid; e < 4096; e += 256) {
      int frag = e >> 6, k = frag >> 2, nj = frag & 3;
      int le = e & 63, lane = le >> 1, v = le & 1;
      int r, c;
      if (lane < 16) { r = k * 4 + v;     c = nj * 16 + lane;      }
      else           { r = k * 4 + 2 + v; c = nj * 16 + lane - 16; }
      wsB[e] = M[r * 64 + c];
    }
  }
}

// ---------------------------------------------------------------------------
// Main: per-wave 64x64 tile, Y = MLT @ (X .* scale) @ MR via f32 WMMA
// ---------------------------------------------------------------------------
#define ROWS 68                 // padded LDS row stride (floats)

__global__ __launch_bounds__(128) void trans_kernel(
    const float* __restrict__ x, const float* __restrict__ dscale,
    const float* __restrict__ wsA, const float* __restrict__ wsB,
    float* __restrict__ out) {
  extern __shared__ float sm[];
  float* fragA = sm;            // 4096: MLT A-fragments
  float* fragB = sm + 4096;     // 4096: MR  B-fragments
  float* bufs  = sm + 8192;     // 4 waves * 64*ROWS
  int tid = threadIdx.x;

  for (int i = tid; i < 1024; i += 128) {
    ((float4*)fragA)[i] = ((const float4*)wsA)[i];
    ((float4*)fragB)[i] = ((const float4*)wsB)[i];
  }

  int wave = tid >> 5, lane = tid & 31;
  float* sb = bufs + wave * (64 * ROWS);
  size_t tile = (size_t)blockIdx.x * 4 + wave;
  const float* xt = x + tile * 4096;
  float*       yt = out + tile * 4096;
  __syncthreads();

  // Stream X tile (b128, coalesced), apply diag_scale, deposit in LDS
#pragma unroll
  for (int i = 0; i < 32; ++i) {
    int f4 = i * 32 + lane;                 // float4 index within tile
    float4 xv = ((const float4*)xt)[f4];
    float4 sv = ((const float4*)dscale)[f4];
    float4 rv;
    rv.x = xv.x * sv.x; rv.y = xv.y * sv.y;
    rv.z = xv.z * sv.z; rv.w = xv.w * sv.w;
    int f = f4 * 4, row = f >> 6, col = f & 63;
    *(float4*)(sb + row * ROWS + col) = rv;
  }
  __builtin_amdgcn_wave_barrier();

  int hi = lane >> 4;            // half-wave select (K/M split)
  int lm = lane & 15;

  // ---- Stage 1: T = MLT @ Xs, overwrite each consumed column block ----
  for (int ni = 0; ni < 4; ++ni) {
    v8f acc[4] = {};
    int c = ni * 16 + lm;
#pragma unroll
    for (int k = 0; k < 16; ++k) {
      int r = k * 4 + hi * 2;
      v2f b;
      b.x = sb[r * ROWS + c];
      b.y = sb[(r + 1) * ROWS + c];
#pragma unroll
      for (int mi = 0; mi < 4; ++mi) {
        v2f a = *(const v2f*)(fragA + (mi * 16 + k) * 64 + lane * 2);
        acc[mi] = __builtin_amdgcn_wmma_f32_16x16x4_f32(
            false, a, false, b, (short)0, acc[mi], false, false);
      }
    }
    __builtin_amdgcn_wave_barrier();
#pragma unroll
    for (int mi = 0; mi < 4; ++mi) {
      int rb = mi * 16 + hi * 8;
#pragma unroll
      for (int j = 0; j < 8; ++j) sb[(rb + j) * ROWS + c] = acc[mi][j];
    }
    __builtin_amdgcn_wave_barrier();
  }

  // ---- Stage 2: Y = T @ MR, store coalesced b32 rows ----
  for (int mi = 0; mi < 4; ++mi) {
    v8f acc[4] = {};
    int ar = mi * 16 + lm;
#pragma unroll
    for (int k = 0; k < 16; ++k) {
      v2f a = *(const v2f*)(sb + ar * ROWS + k * 4 + hi * 2);
#pragma unroll
      for (int nj = 0; nj < 4; ++nj) {
        v2f b = *(const v2f*)(fragB + (k * 4 + nj) * 64 + lane * 2);
        acc[nj] = __builtin_amdgcn_wmma_f32_16x16x4_f32(
            false, a, false, b, (short)0, acc[nj], false, false);
      }
    }
    int rb = mi * 16 + hi * 8;
#pragma unroll
    for (int nj = 0; nj < 4; ++nj)
#pragma unroll
      for (int j = 0; j < 8; ++j)
        yt[(size_t)(rb + j) * 64 + nj * 16 + lm] = acc[nj][j];
  }
}

extern "C" void kernel_launch(void* const* d_in, const int* in_sizes, int n_in,
                              void* d_out, int out_size, void* d_ws, size_t ws_size,
                              hipStream_t stream) {
  const float* x  = (const float*)d_in[0];
  const float* uL = (const float*)d_in[1];
  const float* vL = (const float*)d_in[2];
  const float* dL = (const float*)d_in[3];
  const float* uR = (const float*)d_in[4];
  const float* vR = (const float*)d_in[5];
  const float* dR = (const float*)d_in[6];
  const float* ds = (const float*)d_in[7];
  float* out = (float*)d_out;
  float* wsA = (float*)d_ws;        // 4096 floats: packed MLT A-frags
  float* wsB = wsA + 4096;          // 4096 floats: packed MR B-frags

  size_t setup_lds = (8448 + 4096 + 4096) * sizeof(float);   // ~65 KB
  setup_kernel<<<2, 256, setup_lds, stream>>>(uL, vL, dL, uR, vR, dR, wsA, wsB);

  int n_tiles = in_sizes[0] / 4096;                          // 8192
  size_t main_lds = (4096 + 4096 + 4 * 64 * ROWS) * sizeof(float);  // 100 KB
  trans_kernel<<<n_tiles / 4, 128, main_lds, stream>>>(x, ds, wsA, wsB, out);
}